// get_mmd_loss_44229573214606
// MI455X (gfx1250) — compile-verified
//
#include <hip/hip_runtime.h>
#include <hip/hip_bf16.h>
#include <math.h>

typedef __attribute__((ext_vector_type(2))) float v2f;
typedef __attribute__((ext_vector_type(8))) float v8f;

#define BATCH   4096
#define DFEAT   1024
#define NCLS    40
#define NTOT    8192   // 2*BATCH
#define KDIM    64

__device__ __forceinline__ float wave_reduce_add(float v) {
#pragma unroll
    for (int off = 16; off > 0; off >>= 1) v += __shfl_xor(v, off, 32);
    return v;
}

// ---------------------------------------------------------------------------
// sq[row] = sum_k X[row][k]^2  over virtual concat [dense; sparse]
// ---------------------------------------------------------------------------
__global__ __launch_bounds__(256)
void sq_kernel(const float* __restrict__ dense, const float* __restrict__ sparse,
               float* __restrict__ sq) {
    const int row = blockIdx.x;
    const float* p = (row < BATCH) ? (dense + (size_t)row * DFEAT)
                                   : (sparse + (size_t)(row - BATCH) * DFEAT);
    __shared__ float sm[256];
    float s = 0.f;
    for (int k = threadIdx.x; k < DFEAT; k += 256) { float v = p[k]; s += v * v; }
    sm[threadIdx.x] = s;
    __syncthreads();
    for (int st = 128; st > 0; st >>= 1) {
        if (threadIdx.x < st) sm[threadIdx.x] += sm[threadIdx.x + st];
        __syncthreads();
    }
    if (threadIdx.x == 0) sq[row] = sm[0];
}

// ---------------------------------------------------------------------------
// Main Gram-matrix kernel. PASS==1: accumulate sum(max(l2,0)) per workgroup.
// PASS==2: accumulate 5-Gaussian kernel sums into 4 quadrant partials.
// Workgroup = 4 waves, 128x128 block; each wave: 64x64 via 4x4 WMMA f32 tiles.
// ---------------------------------------------------------------------------
template <int PASS>
__global__ __launch_bounds__(128)
void mmd_gemm(const float* __restrict__ dense, const float* __restrict__ sparse,
              const float* __restrict__ sq, const float* __restrict__ Sptr,
              float* __restrict__ p1_out, float* __restrict__ mmd_out) {
    const int lane = threadIdx.x & 31;
    const int wave = threadIdx.x >> 5;
    const int half = lane >> 4;   // which K pair this lane holds
    const int l    = lane & 15;   // row-within-tile for A/B fragments, N for C

    const int rowBase = blockIdx.y * 128 + (wave >> 1) * 64;
    const int colBase = blockIdx.x * 128 + (wave & 1) * 64;

    // Per-lane row pointers (fixed across the K loop).
    const float* aptr[4];
    const float* bptr[4];
#pragma unroll
    for (int i = 0; i < 4; ++i) {
        const int r = rowBase + i * 16 + l;
        aptr[i] = (r < BATCH) ? (dense + (size_t)r * DFEAT)
                              : (sparse + (size_t)(r - BATCH) * DFEAT);
        const int c = colBase + i * 16 + l;
        bptr[i] = (c < BATCH) ? (dense + (size_t)c * DFEAT)
                              : (sparse + (size_t)(c - BATCH) * DFEAT);
    }

    v8f acc[4][4];
    const v8f zero = {};
#pragma unroll
    for (int i = 0; i < 4; ++i)
#pragma unroll
        for (int j = 0; j < 4; ++j) acc[i][j] = zero;

    for (int k0 = 0; k0 < DFEAT; k0 += 4) {
        const int off = k0 + 2 * half;
        v2f a[4], b[4];
#pragma unroll
        for (int i = 0; i < 4; ++i) a[i] = *(const v2f*)(aptr[i] + off);
#pragma unroll
        for (int j = 0; j < 4; ++j) b[j] = *(const v2f*)(bptr[j] + off);
#pragma unroll
        for (int i = 0; i < 4; ++i)
#pragma unroll
            for (int j = 0; j < 4; ++j)
                acc[i][j] = __builtin_amdgcn_wmma_f32_16x16x4_f32(
                    false, a[i], false, b[j], (short)0, acc[i][j], false, false);
    }

    // Bandwidths (pass 2 only): invbw[i] = 1 / (bw0 * 2^i)
    float invbw0 = 0.f;
    if (PASS == 2) {
        const float S = *Sptr;
        const double n = (double)NTOT;
        const float bw0 = (float)((double)S / (n * n - n)) * 0.25f; // / 2^(5/2 floor)=4
        invbw0 = 1.0f / bw0;
    }

    float lsum = 0.f;
#pragma unroll
    for (int i = 0; i < 4; ++i) {
        const int mbase = rowBase + i * 16 + 8 * half;
        float sqm[8];
#pragma unroll
        for (int r = 0; r < 8; ++r) sqm[r] = sq[mbase + r];
#pragma unroll
        for (int j = 0; j < 4; ++j) {
            const float sqn = sq[colBase + j * 16 + l];
#pragma unroll
            for (int r = 0; r < 8; ++r) {
                float l2 = sqm[r] + sqn - 2.0f * acc[i][j][r];
                l2 = fmaxf(l2, 0.0f);
                if (PASS == 1) {
                    lsum += l2;
                } else {
                    float ib = invbw0;
                    float kv = 0.f;
#pragma unroll
                    for (int q = 0; q < 5; ++q) {
                        kv += __expf(-l2 * ib);
                        ib *= 0.5f;
                    }
                    lsum += kv;
                }
            }
        }
    }

    lsum = wave_reduce_add(lsum);

    __shared__ float wsum[4];
    __shared__ int   wq[4];
    if (lane == 0) {
        wsum[wave] = lsum;
        const int rh = (rowBase >= BATCH) ? 1 : 0;
        const int ch = (colBase >= BATCH) ? 1 : 0;
        wq[wave] = rh * 2 + ch;   // 0=xx 1=xy 2=yx 3=yy
    }
    __syncthreads();
    if (threadIdx.x == 0) {
        const int wg = blockIdx.y * gridDim.x + blockIdx.x;
        if (PASS == 1) {
            p1_out[wg] = wsum[0] + wsum[1] + wsum[2] + wsum[3];
        } else {
            float out4[4] = {0.f, 0.f, 0.f, 0.f};
#pragma unroll
            for (int w = 0; w < 4; ++w) out4[wq[w]] += wsum[w];
#pragma unroll
            for (int c = 0; c < 4; ++c) mmd_out[wg * 4 + c] = out4[c];
        }
    }
}

// ---------------------------------------------------------------------------
// Per-batch ||I - T T^T||_F via WMMA f32. One workgroup (4 waves) per batch.
// Wave w computes rows [16w,16w+16) x 64 cols = 4 tiles, K=64.
// ---------------------------------------------------------------------------
__global__ __launch_bounds__(128)
void frob_kernel(const float* __restrict__ trans, float* __restrict__ out) {
    const int b = blockIdx.x;
    const float* T = trans + (size_t)b * (KDIM * KDIM);
    const int lane = threadIdx.x & 31;
    const int wave = threadIdx.x >> 5;
    const int half = lane >> 4;
    const int l    = lane & 15;
    const int mrow = wave * 16;

    v8f acc[4];
    const v8f zero = {};
#pragma unroll
    for (int j = 0; j < 4; ++j) acc[j] = zero;

    for (int k0 = 0; k0 < KDIM; k0 += 4) {
        const int off = k0 + 2 * half;
        const v2f a = *(const v2f*)(T + (size_t)(mrow + l) * KDIM + off);
#pragma unroll
        for (int j = 0; j < 4; ++j) {
            const v2f bb = *(const v2f*)(T + (size_t)(j * 16 + l) * KDIM + off);
            acc[j] = __builtin_amdgcn_wmma_f32_16x16x4_f32(
                false, a, false, bb, (short)0, acc[j], false, false);
        }
    }

    float s = 0.f;
#pragma unroll
    for (int j = 0; j < 4; ++j) {
        const int n = j * 16 + l;
#pragma unroll
        for (int r = 0; r < 8; ++r) {
            const int m = mrow + 8 * half + r;
            const float d = ((m == n) ? 1.0f : 0.0f) - acc[j][r];
            s += d * d;
        }
    }
    s = wave_reduce_add(s);

    __shared__ float wsum[4];
    if (lane == 0) wsum[wave] = s;
    __syncthreads();
    if (threadIdx.x == 0)
        out[b] = sqrtf(wsum[0] + wsum[1] + wsum[2] + wsum[3]);
}

// ---------------------------------------------------------------------------
// NLL partials: 16 blocks x 256 threads, one element per thread.
// ---------------------------------------------------------------------------
__global__ __launch_bounds__(256)
void nll_kernel(const float* __restrict__ pred, const int* __restrict__ target,
                float* __restrict__ out16) {
    const int i = blockIdx.x * 256 + threadIdx.x;
    __shared__ float sm[256];
    sm[threadIdx.x] = -pred[(size_t)i * NCLS + target[i]];
    __syncthreads();
    for (int st = 128; st > 0; st >>= 1) {
        if (threadIdx.x < st) sm[threadIdx.x] += sm[threadIdx.x + st];
        __syncthreads();
    }
    if (threadIdx.x == 0) out16[blockIdx.x] = sm[0];
}

// ---------------------------------------------------------------------------
// Deterministic single-block reductions.
// ---------------------------------------------------------------------------
__global__ __launch_bounds__(256)
void reduce_sum(const float* __restrict__ in, int n, float* __restrict__ out) {
    __shared__ float sm[256];
    float s = 0.f;
    for (int i = threadIdx.x; i < n; i += 256) s += in[i];
    sm[threadIdx.x] = s;
    __syncthreads();
    for (int st = 128; st > 0; st >>= 1) {
        if (threadIdx.x < st) sm[threadIdx.x] += sm[threadIdx.x + st];
        __syncthreads();
    }
    if (threadIdx.x == 0) *out = sm[0];
}

__global__ __launch_bounds__(256)
void reduce_mmd(const float* __restrict__ in, int nwg, float* __restrict__ quad) {
    __shared__ float sm[256];
    for (int c = 0; c < 4; ++c) {
        float s = 0.f;
        for (int w = threadIdx.x; w < nwg; w += 256) s += in[w * 4 + c];
        sm[threadIdx.x] = s;
        __syncthreads();
        for (int st = 128; st > 0; st >>= 1) {
            if (threadIdx.x < st) sm[threadIdx.x] += sm[threadIdx.x + st];
            __syncthreads();
        }
        if (threadIdx.x == 0) quad[c] = sm[0];
        __syncthreads();
    }
}

// ---------------------------------------------------------------------------
// Final combine.
// ---------------------------------------------------------------------------
__global__ void final_kernel(const float* __restrict__ nll16,
                             const float* __restrict__ frobsum,
                             const float* __restrict__ quad,
                             float* __restrict__ out) {
    if (threadIdx.x == 0 && blockIdx.x == 0) {
        float nll = 0.f;
        for (int i = 0; i < 16; ++i) nll += nll16[i];
        nll *= (1.0f / (float)BATCH);
        const float mat = (*frobsum) * (1.0f / (float)BATCH);
        const float mmd = (quad[0] + quad[3] - quad[1] - quad[2]) *
                          (1.0f / ((float)BATCH * (float)BATCH));
        out[0] = 0.1f * nll + 0.001f * mat + 0.5f * mmd;
    }
}

// ---------------------------------------------------------------------------
extern "C" void kernel_launch(void* const* d_in, const int* in_sizes, int n_in,
                              void* d_out, int out_size, void* d_ws, size_t ws_size,
                              hipStream_t stream) {
    const float* pred       = (const float*)d_in[0];
    const int*   target     = (const int*)d_in[1];
    const float* trans_feat = (const float*)d_in[2];
    const float* dense      = (const float*)d_in[3];
    const float* sparse     = (const float*)d_in[4];
    float* out = (float*)d_out;

    float* w       = (float*)d_ws;
    float* wsS     = w + 0;          // 1: sum(l2)
    float* wsFrobS = w + 1;          // 1: sum of frob norms
    float* wsQuad  = w + 4;          // 4: xx, xy, yx, yy sums
    float* wsNll   = w + 8;          // 16: nll partials
    float* wsSq    = w + 32;         // 8192: row squared norms
    float* wsP1    = w + 8224;       // 4096: pass-1 partials
    float* wsMmd   = w + 12320;      // 16384: pass-2 partials (4 per wg)
    float* wsFrob  = w + 28704;      // 4096: per-batch frob norms

    (void)in_sizes; (void)n_in; (void)out_size; (void)ws_size;

    // 1) squared row norms of [dense; sparse]
    sq_kernel<<<NTOT, 256, 0, stream>>>(dense, sparse, wsSq);

    // 2) pass-1 GEMM: sum of clamped pairwise distances (for bandwidth)
    dim3 gemmGrid(NTOT / 128, NTOT / 128);
    mmd_gemm<1><<<gemmGrid, 128, 0, stream>>>(dense, sparse, wsSq, wsS, wsP1, wsMmd);
    reduce_sum<<<1, 256, 0, stream>>>(wsP1, 4096, wsS);

    // 3) pass-2 GEMM: 5-Gaussian kernel quadrant sums
    mmd_gemm<2><<<gemmGrid, 128, 0, stream>>>(dense, sparse, wsSq, wsS, wsP1, wsMmd);
    reduce_mmd<<<1, 256, 0, stream>>>(wsMmd, 4096, wsQuad);

    // 4) regularizer: per-batch ||I - T T^T||_F, then mean
    frob_kernel<<<BATCH, 128, 0, stream>>>(trans_feat, wsFrob);
    reduce_sum<<<1, 256, 0, stream>>>(wsFrob, BATCH, wsFrobS);

    // 5) NLL partials
    nll_kernel<<<16, 256, 0, stream>>>(pred, target, wsNll);

    // 6) combine
    final_kernel<<<1, 32, 0, stream>>>(wsNll, wsFrobS, wsQuad, out);
}